// ResidualBlock_3135326126714
// MI455X (gfx1250) — compile-verified
//
#include <hip/hip_runtime.h>
#include <math.h>

// ---------------------------------------------------------------------------
// GMMConv residual block for MI455X (gfx1250, wave32).
// fp32 throughout (memory/L2 bound problem; no bandwidth win from fp16).
// Dense projections use V_WMMA_F32_16X16X4_F32.
// ---------------------------------------------------------------------------

typedef float v2f __attribute__((ext_vector_type(2)));
typedef float v8f __attribute__((ext_vector_type(8)));

#define GMM_EPS 1e-15f
#define BN_EPS  1e-5f

// ---------------------------------------------------------------------------
// Zero a float buffer
__global__ void zero_f32_kernel(float* __restrict__ p, long n) {
    long i = (long)blockIdx.x * blockDim.x + threadIdx.x;
    if (i < n) p[i] = 0.f;
}

// ---------------------------------------------------------------------------
// Pack weights: wpack[32][256] = [g1 (32x160) | root1 (32x32) | gs (32x32) | roots (32x32)]
__global__ void pack1_kernel(const float* __restrict__ g1, const float* __restrict__ r1,
                             const float* __restrict__ gs, const float* __restrict__ rs,
                             float* __restrict__ wp) {
    int i = blockIdx.x * blockDim.x + threadIdx.x;
    if (i >= 32 * 256) return;
    int k = i >> 8, c = i & 255;
    float v;
    if      (c < 160) v = g1[k * 160 + c];
    else if (c < 192) v = r1[k * 32 + (c - 160)];
    else if (c < 224) v = gs[k * 32 + (c - 192)];
    else              v = rs[k * 32 + (c - 224)];
    wp[i] = v;
}

// wpack[32][192] = [g2 (32x160) | root2 (32x32)]
__global__ void pack2_kernel(const float* __restrict__ g2, const float* __restrict__ r2,
                             float* __restrict__ wp) {
    int i = blockIdx.x * blockDim.x + threadIdx.x;
    if (i >= 32 * 192) return;
    int k = i / 192, c = i - k * 192;
    wp[i] = (c < 160) ? g2[k * 160 + c] : r2[k * 32 + (c - 160)];
}

// ---------------------------------------------------------------------------
// WMMA GEMM: Out[n x C] = A[n x 32] @ Bp[32 x C], C <= 256, C multiple of 16.
// Block: 512 threads = 16 waves; each wave owns one 16-wide column tile and
// iterates 4 row tiles (64 rows per block). K=32 -> 8 x V_WMMA_F32_16X16X4_F32.
//
// f32 WMMA VGPR layouts (ISA 7.12.2):
//   A (16x4): lanes 0-15 rows M=0..15; vgpr0 = K0 (lo lanes)/K2 (hi), vgpr1 = K1/K3
//   B (4x16): mirrored: lanes hold N=0..15; vgpr0 = K0/K2, vgpr1 = K1/K3
//   C/D (16x16): vgpr j = row j (lanes 0-15) / row j+8 (lanes 16-31), col = lane&15
__global__ __launch_bounds__(512)
void gemm32_wmma_kernel(const float* __restrict__ A, const float* __restrict__ Bp,
                        float* __restrict__ Out, int n, int C) {
    __shared__ float As[64][32];
    __shared__ float Bs[32 * 256];

    int tid = threadIdx.x;
    int rowBase = blockIdx.x * 64;

    // Stage B (32 x C) and A (64 x 32) tiles in LDS.
    int btot = 32 * C;
    for (int i = tid; i < btot; i += 512) Bs[i] = Bp[i];
    for (int i = tid; i < 64 * 32; i += 512) {
        int r = i >> 5, k = i & 31;
        int gr = rowBase + r;
        As[r][k] = (gr < n) ? A[(size_t)gr * 32 + k] : 0.f;
    }
    __syncthreads();

    int wave = tid >> 5;
    int lane = tid & 31;
    int c0 = wave * 16;
    if (c0 >= C) return;            // wave-uniform: EXEC stays all-ones for WMMA

    int nidx = lane & 15;           // col within tile; also A row (M) for this lane
    int hi   = lane >> 4;           // lane group selects K pair {0,1} vs {2,3}

    for (int t = 0; t < 4; ++t) {
        v8f acc = {};
        int arow = t * 16 + nidx;
#pragma unroll
        for (int k0 = 0; k0 < 32; k0 += 4) {
            int ka = k0 + 2 * hi;
            v2f a, b;
            a.x = As[arow][ka];
            a.y = As[arow][ka + 1];
            b.x = Bs[ka * C + c0 + nidx];
            b.y = Bs[(ka + 1) * C + c0 + nidx];
            // 8 args: (neg_a, A, neg_b, B, c_mod, C, reuse_a, reuse_b)
            acc = __builtin_amdgcn_wmma_f32_16x16x4_f32(
                false, a, false, b, (short)0, acc, false, false);
        }
#pragma unroll
        for (int j = 0; j < 8; ++j) {
            int grow = rowBase + t * 16 + j + hi * 8;
            if (grow < n) Out[(size_t)grow * C + c0 + nidx] = acc[j];
        }
    }
}

// ---------------------------------------------------------------------------
// Gaussian kernel weight: exp(-0.5 * sum_d (a_d - mu_kd)^2 / (eps + sigma_kd^2))
__device__ __forceinline__ float gweight(const float* __restrict__ mu,
                                         const float* __restrict__ sg, int k,
                                         float a0, float a1, float a2) {
    float d0 = a0 - mu[k * 3 + 0], d1 = a1 - mu[k * 3 + 1], d2 = a2 - mu[k * 3 + 2];
    float s0 = sg[k * 3 + 0],      s1 = sg[k * 3 + 1],      s2 = sg[k * 3 + 2];
    float e = d0 * d0 / (GMM_EPS + s0 * s0)
            + d1 * d1 / (GMM_EPS + s1 * s1)
            + d2 * d2 / (GMM_EPS + s2 * s2);
    return __expf(-0.5f * e);
}

// ---------------------------------------------------------------------------
// Edge pass 1: fuses conv1 (K=5) and shortcut conv (K=1) message+scatter.
// One wave per edge, lane = output channel. Gathers hit the N*256*4B = 102MB
// xcat table, which lives in the 192MB L2.
__global__ void edge_pass1_kernel(const int* __restrict__ ei, const float* __restrict__ ea,
                                  const float* __restrict__ xcat,
                                  const float* __restrict__ mu1, const float* __restrict__ sg1,
                                  const float* __restrict__ mus, const float* __restrict__ sgs,
                                  float* __restrict__ agg1, float* __restrict__ aggs,
                                  float* __restrict__ cnt, int E) {
    int g = blockIdx.x * blockDim.x + threadIdx.x;
    int e = g >> 5;
    if (e >= E) return;
    int lane = g & 31;
    int src = ei[e];
    int dst = ei[E + e];
    float a0 = ea[e * 3 + 0], a1 = ea[e * 3 + 1], a2 = ea[e * 3 + 2];

    const float* xb = xcat + (size_t)src * 256;
    float m1 = 0.f;
#pragma unroll
    for (int k = 0; k < 5; ++k) {
        float w = gweight(mu1, sg1, k, a0, a1, a2);
        m1 += w * xb[k * 32 + lane];
    }
    float wsv = gweight(mus, sgs, 0, a0, a1, a2);
    float ms  = wsv * xb[192 + lane];

    atomicAdd(&agg1[(size_t)dst * 32 + lane], m1);
    atomicAdd(&aggs[(size_t)dst * 32 + lane], ms);
    if (lane == 0) atomicAdd(&cnt[dst], 1.f);
}

// Edge pass 2: conv2 (K=5) over h-projections (stride 192).
__global__ void edge_pass2_kernel(const int* __restrict__ ei, const float* __restrict__ ea,
                                  const float* __restrict__ hcat,
                                  const float* __restrict__ mu2, const float* __restrict__ sg2,
                                  float* __restrict__ agg2, int E) {
    int g = blockIdx.x * blockDim.x + threadIdx.x;
    int e = g >> 5;
    if (e >= E) return;
    int lane = g & 31;
    int src = ei[e];
    int dst = ei[E + e];
    float a0 = ea[e * 3 + 0], a1 = ea[e * 3 + 1], a2 = ea[e * 3 + 2];

    const float* xb = hcat + (size_t)src * 192;
    float m = 0.f;
#pragma unroll
    for (int k = 0; k < 5; ++k) {
        float w = gweight(mu2, sg2, k, a0, a1, a2);
        m += w * xb[k * 32 + lane];
    }
    atomicAdd(&agg2[(size_t)dst * 32 + lane], m);
}

// ---------------------------------------------------------------------------
// Stage 1: y1 = agg1/cnt + x@root1 + b1 (in-place aggA),
//          ys = aggs/cnt + x@roots + bs (in-place aggB),
// accumulate BN sums/sumsqs (sets 0 and 1) via LDS ds_add_f32 then global atomics.
// stats layout: [set*64 + c] = sum, [set*64 + 32 + c] = sumsq, sets 0/1/2.
// finalized (mean, rstd) at stats+192 with same layout.
__global__ void stage1_kernel(float* __restrict__ aggA, float* __restrict__ aggB,
                              const float* __restrict__ cnt, const float* __restrict__ xcat,
                              const float* __restrict__ b1, const float* __restrict__ bs,
                              float* __restrict__ stats, int N) {
    __shared__ float s[4][32];
    int tid = threadIdx.x;
    if (tid < 128) ((float*)s)[tid] = 0.f;
    __syncthreads();
    int gid = blockIdx.x * blockDim.x + tid;
    if (gid < N * 32) {
        int node = gid >> 5, c = gid & 31;
        float invc = 1.f / fmaxf(cnt[node], 1.f);
        const float* xr = xcat + (size_t)node * 256;
        float y1 = aggA[gid] * invc + xr[160 + c] + b1[c];
        float ys = aggB[gid] * invc + xr[224 + c] + bs[c];
        aggA[gid] = y1;
        aggB[gid] = ys;
        atomicAdd(&s[0][c], y1); atomicAdd(&s[1][c], y1 * y1);
        atomicAdd(&s[2][c], ys); atomicAdd(&s[3][c], ys * ys);
    }
    __syncthreads();
    if (tid < 128) atomicAdd(&stats[tid], ((float*)s)[tid]);
}

// Stage 2: y2 = agg2/cnt + h@root2 + b2 (in-place aggC), BN stats set 2.
__global__ void stage2_kernel(float* __restrict__ aggC, const float* __restrict__ cnt,
                              const float* __restrict__ hcat, const float* __restrict__ b2,
                              float* __restrict__ stats, int N) {
    __shared__ float s[2][32];
    int tid = threadIdx.x;
    if (tid < 64) ((float*)s)[tid] = 0.f;
    __syncthreads();
    int gid = blockIdx.x * blockDim.x + tid;
    if (gid < N * 32) {
        int node = gid >> 5, c = gid & 31;
        float invc = 1.f / fmaxf(cnt[node], 1.f);
        float y2 = aggC[gid] * invc + hcat[(size_t)node * 192 + 160 + c] + b2[c];
        aggC[gid] = y2;
        atomicAdd(&s[0][c], y2); atomicAdd(&s[1][c], y2 * y2);
    }
    __syncthreads();
    if (tid < 64) atomicAdd(&stats[128 + tid], ((float*)s)[tid]);
}

// Finalize BN stats for [set0, set0+nsets): mean + rstd.
__global__ void finalize_kernel(float* __restrict__ stats, int set0, int nsets, float invN) {
    int t = threadIdx.x;
    if (t >= nsets * 32) return;
    int set = set0 + (t >> 5), c = t & 31;
    float m = stats[set * 64 + c] * invN;
    float v = stats[set * 64 + 32 + c] * invN - m * m;
    stats[192 + set * 64 + c] = m;
    stats[192 + set * 64 + 32 + c] = rsqrtf(v + BN_EPS);
}

// h = elu(bn1(y1)), in-place.
__global__ void bn_elu_kernel(float* __restrict__ y, const float* __restrict__ stats,
                              const float* __restrict__ gam, const float* __restrict__ bet,
                              int n32) {
    int i = blockIdx.x * blockDim.x + threadIdx.x;
    if (i >= n32) return;
    int c = i & 31;
    float m = stats[192 + c], r = stats[192 + 32 + c];
    float h = (y[i] - m) * r * gam[c] + bet[c];
    y[i] = h > 0.f ? h : expm1f(h);
}

// out = elu( bn2(y2) + bns(ys) )
__global__ void final_kernel(const float* __restrict__ y2v, const float* __restrict__ ysv,
                             const float* __restrict__ stats,
                             const float* __restrict__ gam2, const float* __restrict__ bet2,
                             const float* __restrict__ gams, const float* __restrict__ bets,
                             float* __restrict__ out, int n32) {
    int i = blockIdx.x * blockDim.x + threadIdx.x;
    if (i >= n32) return;
    int c = i & 31;
    const float* F = stats + 192;
    float h = (y2v[i] - F[128 + c]) * F[160 + c] * gam2[c] + bet2[c];
    float s = (ysv[i] - F[64 + c])  * F[96 + c]  * gams[c] + bets[c];
    float o = h + s;
    out[i] = o > 0.f ? o : expm1f(o);
}

// ---------------------------------------------------------------------------
extern "C" void kernel_launch(void* const* d_in, const int* in_sizes, int n_in,
                              void* d_out, int out_size, void* d_ws, size_t ws_size,
                              hipStream_t stream) {
    const float* x     = (const float*)d_in[0];
    const int*   ei    = (const int*)  d_in[1];
    const float* ea    = (const float*)d_in[2];
    const float* g1    = (const float*)d_in[3];
    const float* mu1   = (const float*)d_in[4];
    const float* sg1   = (const float*)d_in[5];
    const float* root1 = (const float*)d_in[6];
    const float* b1    = (const float*)d_in[7];
    const float* gam1  = (const float*)d_in[8];
    const float* bet1  = (const float*)d_in[9];
    const float* g2    = (const float*)d_in[10];
    const float* mu2   = (const float*)d_in[11];
    const float* sg2   = (const float*)d_in[12];
    const float* root2 = (const float*)d_in[13];
    const float* b2    = (const float*)d_in[14];
    const float* gam2  = (const float*)d_in[15];
    const float* bet2  = (const float*)d_in[16];
    const float* gs    = (const float*)d_in[17];
    const float* mus   = (const float*)d_in[18];
    const float* sgs   = (const float*)d_in[19];
    const float* roots = (const float*)d_in[20];
    const float* bs    = (const float*)d_in[21];
    const float* gams  = (const float*)d_in[22];
    const float* bets  = (const float*)d_in[23];

    const int N = in_sizes[0] / 32;
    const int E = in_sizes[1] / 2;

    // workspace layout (floats)
    float* w     = (float*)d_ws;
    float* xcat  = w;                          // N*256  (x@[g1|root1|gs|roots], later h@[g2|root2])
    float* aggA  = xcat + (size_t)N * 256;     // N*32   (agg1 -> y1 -> h)
    float* aggB  = aggA + (size_t)N * 32;      // N*32   (aggs -> ys)
    float* aggC  = aggB + (size_t)N * 32;      // N*32   (agg2 -> y2)
    float* cnt   = aggC + (size_t)N * 32;      // N
    float* wpack = cnt + (size_t)N;            // 32*256
    float* stats = wpack + 32 * 256;           // 384

    const long zlen = (long)N * 97 + 32 * 256 + 384;   // aggA..stats inclusive
    const int n32   = N * 32;
    const int TB    = 256;

    zero_f32_kernel<<<(int)((zlen + TB - 1) / TB), TB, 0, stream>>>(aggA, zlen);

    // projections of x: xcat = x @ [g1 | root1 | gs | roots]
    pack1_kernel<<<(32 * 256 + TB - 1) / TB, TB, 0, stream>>>(g1, root1, gs, roots, wpack);
    gemm32_wmma_kernel<<<(N + 63) / 64, 512, 0, stream>>>(x, wpack, xcat, N, 256);

    // fused conv1 + shortcut edge pass
    int egrid = (int)(((long)E * 32 + TB - 1) / TB);
    edge_pass1_kernel<<<egrid, TB, 0, stream>>>(ei, ea, xcat, mu1, sg1, mus, sgs,
                                                aggA, aggB, cnt, E);

    stage1_kernel<<<(n32 + TB - 1) / TB, TB, 0, stream>>>(aggA, aggB, cnt, xcat, b1, bs, stats, N);
    finalize_kernel<<<1, 64, 0, stream>>>(stats, 0, 2, 1.f / (float)N);
    bn_elu_kernel<<<(n32 + TB - 1) / TB, TB, 0, stream>>>(aggA, stats, gam1, bet1, n32);

    // projections of h: xcat = h @ [g2 | root2]  (reuses xcat region, stride 192)
    pack2_kernel<<<(32 * 192 + TB - 1) / TB, TB, 0, stream>>>(g2, root2, wpack);
    gemm32_wmma_kernel<<<(N + 63) / 64, 512, 0, stream>>>(aggA, wpack, xcat, N, 192);

    edge_pass2_kernel<<<egrid, TB, 0, stream>>>(ei, ea, xcat, mu2, sg2, aggC, E);

    stage2_kernel<<<(n32 + TB - 1) / TB, TB, 0, stream>>>(aggC, cnt, xcat, b2, stats, N);
    finalize_kernel<<<1, 32, 0, stream>>>(stats, 2, 1, 1.f / (float)N);

    final_kernel<<<(n32 + TB - 1) / TB, TB, 0, stream>>>(aggC, aggB, stats,
                                                         gam2, bet2, gams, bets,
                                                         (float*)d_out, n32);
}